// RefinedQuantumEntanglementLayer_48799418417351
// MI455X (gfx1250) — compile-verified
//
#include <hip/hip_runtime.h>
#include <math.h>

// Problem sizes (fixed by the reference).
#define S_DIM 2048
#define B_DIM 4096

// expm(A) = (TaylorN(A/2^k))^(2^k); ||A/32|| ~ 0.4 -> 8 Taylor terms ~1e-9.
#define EXPM_SQUARINGS 5
#define EXPM_SCALE     (1.0f / 32.0f)
#define EXPM_TERMS     8

typedef __attribute__((ext_vector_type(2))) float v2f;
typedef __attribute__((ext_vector_type(8))) float v8f;

// ---------------------------------------------------------------------------
// Tiled f32 GEMM on CDNA5 WMMA: C = epilogue(scale * A[MxK] @ B[KxN])
//   MODE 0: C = scale * (A@B)
//   MODE 1: C = scale * (A@B) + I              (Taylor/Horner step)
//   MODE 2: C = (A@B + bias[n])^2 * gp[n]      (quantum measure + gate)
//
// Block tile 128x128, K-stage 32, 8 waves as 4x2; each wave owns 32x64
// (2x4 tiles of v_wmma_f32_16x16x4_f32, 64 accumulator VGPRs).
//
// Global->LDS staging is software-pipelined through 32 dedicated VGPRs:
// current tile's registers are dumped to LDS, then the NEXT tile's 8
// global_load_b128 are issued immediately (no wait) before the WMMA loop,
// so load latency hides behind 64 WMMAs per stage.
//
// LDS layouts (wave32 f32 WMMA operand layouts):
//  - A: row-major, pitch 36. Lane reads row lane%16, K-pair 2*(lane/16):
//    one aligned 8B ds_load per fragment; banks 4*(9r mod 16) conflict-free.
//  - B: K-PAIR INTERLEAVED: element (k,n) at [(k>>1)*288 + 2n + (k&1)], so a
//    lane's {B[k][n],B[k+1][n]} is ONE aligned ds_load_b64 directly into the
//    WMMA operand pair (no v_mov repacking). 288 mod 64 == 32 puts the two
//    K-half lane groups on disjoint bank ranges.
// ---------------------------------------------------------------------------
#define BM 128
#define BN 128
#define BK 32
#define LDA_S 36        // BK + 4
#define LDBP  288       // 2*BN + 32

template <int MODE>
__global__ __launch_bounds__(256) void gemm_wmma_f32(
    const float* __restrict__ A, const float* __restrict__ Bm,
    float* __restrict__ C, int M, int N, int K, float scale,
    const float* __restrict__ bias, const float* __restrict__ gp) {
  __shared__ float sA[BM * LDA_S];           // 128*36*4 = 18432 B
  __shared__ float sB[(BK / 2) * LDBP];      // 16*288*4 = 18432 B

  const int tid  = threadIdx.x;
  const int lane = tid & 31;
  const int wave = tid >> 5;
  const int wr   = wave >> 1;  // 0..3  (wave row)
  const int wc   = wave & 1;   // 0..1  (wave col)
  const int half = lane >> 4;  // selects K pair (wave32 f32 WMMA layout)
  const int lr   = lane & 15;  // row (A) / col (B) within a 16-tile

  const int blockRow = blockIdx.y * BM;
  const int blockCol = blockIdx.x * BN;
  const int wrow0 = wr * 32;   // 32 rows per wave
  const int wcol0 = wc * 64;   // 64 cols per wave

  // Per-thread staging coordinates (4 float4 for A, 4 for B).
  int arow[4], acol[4], brow[4], bcol[4];
#pragma unroll
  for (int i = 0; i < 4; ++i) {
    const int idx = tid + i * 256;
    arow[i] = idx >> 3;          // A: 32 floats/row -> 8 f4 per row
    acol[i] = (idx & 7) << 2;
    brow[i] = idx >> 5;          // B: 128 floats/row -> 32 f4 per row
    bcol[i] = (idx & 31) << 2;
  }

  float4 pa[4], pb[4];
  auto load_tiles = [&](int kb) {
#pragma unroll
    for (int i = 0; i < 4; ++i)
      pa[i] = *(const float4*)&A[(size_t)(blockRow + arow[i]) * K + kb + acol[i]];
#pragma unroll
    for (int i = 0; i < 4; ++i)
      pb[i] = *(const float4*)&Bm[(size_t)(kb + brow[i]) * N + blockCol + bcol[i]];
  };

  v8f acc[2][4];
#pragma unroll
  for (int i = 0; i < 2; ++i)
#pragma unroll
    for (int j = 0; j < 4; ++j)
#pragma unroll
      for (int v = 0; v < 8; ++v) acc[i][j][v] = 0.0f;

  load_tiles(0);  // prologue: first tile in flight

  for (int kb = 0; kb < K; kb += BK) {
    // Dump staged registers to LDS (waits on the in-flight loads here).
#pragma unroll
    for (int i = 0; i < 4; ++i) {
      float* dst = &sA[arow[i] * LDA_S + acol[i]];
      dst[0] = pa[i].x; dst[1] = pa[i].y; dst[2] = pa[i].z; dst[3] = pa[i].w;
    }
#pragma unroll
    for (int i = 0; i < 4; ++i) {
      float* dst = &sB[(brow[i] >> 1) * LDBP + 2 * bcol[i] + (brow[i] & 1)];
      dst[0] = pb[i].x; dst[2] = pb[i].y; dst[4] = pb[i].z; dst[6] = pb[i].w;
    }
    __syncthreads();

    // Issue next tile's global loads NOW; latency hides behind the WMMAs.
    if (kb + BK < K) load_tiles(kb + BK);

#pragma unroll
    for (int kk = 0; kk < BK; kk += 4) {
      // A fragment 16x4: lane = row lr, VGPR0/1 = K = kk+2*half, +1.
      v2f a[2];
#pragma unroll
      for (int i = 0; i < 2; ++i)
        a[i] = *(const v2f*)&sA[(wrow0 + i * 16 + lr) * LDA_S + kk + 2 * half];
      // B fragment 4x16: lane = col lr; one b64 load = {B[k][n],B[k+1][n]}.
      const int kkp = (kk >> 1) + half;
      v2f b[4];
#pragma unroll
      for (int j = 0; j < 4; ++j)
        b[j] = *(const v2f*)&sB[kkp * LDBP + 2 * (wcol0 + j * 16 + lr)];
#pragma unroll
      for (int i = 0; i < 2; ++i)
#pragma unroll
        for (int j = 0; j < 4; ++j)
          acc[i][j] = __builtin_amdgcn_wmma_f32_16x16x4_f32(
              false, a[i], false, b[j], (short)0, acc[i][j], false, false);
    }
    __syncthreads();
  }

  // Epilogue. C/D layout: VGPR v, lanes 0-15 -> M=v, lanes 16-31 -> M=v+8.
#pragma unroll
  for (int i = 0; i < 2; ++i) {
#pragma unroll
    for (int j = 0; j < 4; ++j) {
#pragma unroll
      for (int v = 0; v < 8; ++v) {
        const int gr = blockRow + wrow0 + i * 16 + v + 8 * half;
        const int gc = blockCol + wcol0 + j * 16 + lr;
        const float val = acc[i][j][v];
        if (MODE == 0) {
          C[(size_t)gr * N + gc] = scale * val;
        } else if (MODE == 1) {
          C[(size_t)gr * N + gc] = scale * val + ((gr == gc) ? 1.0f : 0.0f);
        } else {
          const float t = val + bias[gc];
          C[(size_t)gr * N + gc] = t * t * gp[gc];
        }
      }
    }
  }
}

// ---------------------------------------------------------------------------
// Small helper kernels
// ---------------------------------------------------------------------------
__global__ void skew_scale(const float* __restrict__ M, float* __restrict__ out,
                           int n, float s) {
  const int j = blockIdx.x * 16 + threadIdx.x;
  const int i = blockIdx.y * 16 + threadIdx.y;
  out[(size_t)i * n + j] = (M[(size_t)i * n + j] - M[(size_t)j * n + i]) * s;
}

__global__ void set_identity(float* __restrict__ P, int n) {
  const int j = blockIdx.x * 16 + threadIdx.x;
  const int i = blockIdx.y * 16 + threadIdx.y;
  P[(size_t)i * n + j] = (i == j) ? 1.0f : 0.0f;
}

__global__ __launch_bounds__(256) void l2_normalize_rows(
    const float* __restrict__ X, float* __restrict__ Y, int ncols) {
  __shared__ float red[256];
  const int row = blockIdx.x;
  const float* xr = X + (size_t)row * ncols;
  float s = 0.0f;
  for (int c = threadIdx.x; c < ncols; c += 256) {
    const float v = xr[c];
    s += v * v;
  }
  red[threadIdx.x] = s;
  __syncthreads();
  for (int off = 128; off > 0; off >>= 1) {
    if (threadIdx.x < off) red[threadIdx.x] += red[threadIdx.x + off];
    __syncthreads();
  }
  const float inv = 1.0f / sqrtf(fmaxf(red[0], 1e-12f));
  float* yr = Y + (size_t)row * ncols;
  for (int c = threadIdx.x; c < ncols; c += 256) yr[c] = xr[c] * inv;
}

__global__ __launch_bounds__(256) void softmax_vec(
    const float* __restrict__ x, float* __restrict__ out1,
    float* __restrict__ out2, int n) {
  __shared__ float red[256];
  float m = -3.402823466e38f;
  for (int c = threadIdx.x; c < n; c += 256) m = fmaxf(m, x[c]);
  red[threadIdx.x] = m;
  __syncthreads();
  for (int off = 128; off > 0; off >>= 1) {
    if (threadIdx.x < off)
      red[threadIdx.x] = fmaxf(red[threadIdx.x], red[threadIdx.x + off]);
    __syncthreads();
  }
  const float mx = red[0];
  __syncthreads();
  float s = 0.0f;
  for (int c = threadIdx.x; c < n; c += 256) s += expf(x[c] - mx);
  red[threadIdx.x] = s;
  __syncthreads();
  for (int off = 128; off > 0; off >>= 1) {
    if (threadIdx.x < off) red[threadIdx.x] += red[threadIdx.x + off];
    __syncthreads();
  }
  const float inv = 1.0f / red[0];
  for (int c = threadIdx.x; c < n; c += 256) {
    const float p = expf(x[c] - mx) * inv;
    out1[c] = p;
    out2[c] = p;
  }
}

// ---------------------------------------------------------------------------
// Host-side launch helpers (all stream-ordered; graph-capture safe)
// ---------------------------------------------------------------------------
static inline void launch_gemm(int mode, const float* A, const float* Bm,
                               float* C, int M, int N, int K, float scale,
                               const float* bias, const float* gp,
                               hipStream_t st) {
  dim3 grid(N / BN, M / BM), block(256);
  if (mode == 0)
    gemm_wmma_f32<0><<<grid, block, 0, st>>>(A, Bm, C, M, N, K, scale, bias, gp);
  else if (mode == 1)
    gemm_wmma_f32<1><<<grid, block, 0, st>>>(A, Bm, C, M, N, K, scale, bias, gp);
  else
    gemm_wmma_f32<2><<<grid, block, 0, st>>>(A, Bm, C, M, N, K, scale, bias, gp);
}

// expm of skew-symmetrized Msrc: scaling (2^-5), 8-term Horner Taylor,
// 5 repeated squarings. Returns pointer (X0 or X1) holding the result.
static float* run_expm(const float* Msrc, float* As, float* X0, float* X1,
                       hipStream_t st) {
  dim3 g2(S_DIM / 16, S_DIM / 16), b2(16, 16);
  skew_scale<<<g2, b2, 0, st>>>(Msrc, As, S_DIM, EXPM_SCALE);
  set_identity<<<g2, b2, 0, st>>>(X0, S_DIM);
  float* P = X0;
  float* T = X1;
  for (int j = EXPM_TERMS; j >= 1; --j) {  // P <- I + (As @ P) / j
    launch_gemm(1, As, P, T, S_DIM, S_DIM, S_DIM, 1.0f / (float)j, nullptr,
                nullptr, st);
    float* tmp = P; P = T; T = tmp;
  }
  for (int i = 0; i < EXPM_SQUARINGS; ++i) {  // P <- P @ P
    launch_gemm(0, P, P, T, S_DIM, S_DIM, S_DIM, 1.0f, nullptr, nullptr, st);
    float* tmp = P; P = T; T = tmp;
  }
  return P;
}

extern "C" void kernel_launch(void* const* d_in, const int* in_sizes, int n_in,
                              void* d_out, int out_size, void* d_ws,
                              size_t ws_size, hipStream_t stream) {
  (void)in_sizes; (void)n_in; (void)out_size; (void)ws_size;
  const float* inputs     = (const float*)d_in[0];  // [B,S]
  const float* M_real     = (const float*)d_in[1];  // [S,S]
  const float* M_imag     = (const float*)d_in[2];  // [S,S]
  const float* bias       = (const float*)d_in[3];  // [S]
  const float* qubit_gate = (const float*)d_in[4];  // [S]

  const size_t SS = (size_t)S_DIM * S_DIM;
  float* W0 = (float*)d_ws;                      // S*S : skew gen / final U
  float* W1 = W0 + SS;                           // S*S : ping-pong
  float* W2 = W1 + SS;                           // S*S : ping-pong
  float* W3 = W2 + SS;                           // S*S : ping-pong (expm #2)
  float* NI = W3 + SS;                           // B*S : normalized inputs
  float* GP = NI + (size_t)B_DIM * S_DIM;        // S   : gate probs

  float* out    = (float*)d_out;                 // [B,S] gated output
  float* gp_out = out + (size_t)B_DIM * S_DIM;   // [S]   gate probs output

  // Cheap prologue (independent of expm chain).
  softmax_vec<<<1, 256, 0, stream>>>(qubit_gate, GP, gp_out, S_DIM);
  l2_normalize_rows<<<B_DIM, 256, 0, stream>>>(inputs, NI, S_DIM);

  // E1 = expm(M_real - M_real^T)   (result lands in W1 or W2)
  float* E1 = run_expm(M_real, W0, W1, W2, stream);
  float* free1 = (E1 == W1) ? W2 : W1;
  // E2 = expm(M_imag - M_imag^T)   (reuses W0 for the generator)
  float* E2 = run_expm(M_imag, W0, W3, free1, stream);

  // U = E1 @ E2  (W0 is free again after the second Horner chain)
  float* U = W0;
  launch_gemm(0, E1, E2, U, S_DIM, S_DIM, S_DIM, 1.0f, nullptr, nullptr,
              stream);

  // out = (NI @ U + bias)^2 * gate_probs   (fused epilogue)
  launch_gemm(2, NI, U, out, B_DIM, S_DIM, S_DIM, 1.0f, bias, GP, stream);
}